// LocalSpatialEncoder_26388279067325
// MI455X (gfx1250) — compile-verified
//
#include <hip/hip_runtime.h>

// ChebNet graph conv, fully fused on MI455X (gfx1250).
//   Z_k = 2 L Z_{k-1} - Z_{k-2};  out = sum_k Z_k W_k + bias
// ~128 GFLOP fp32 vs ~288 MB HBM traffic (AI ~440 F/B) -> compute bound, so:
//  * read x once, keep the whole recurrence in LDS (no Z to HBM),
//  * every MAC through V_WMMA_F32_16X16X4_F32 (fp32 == reference precision),
//  * L (412 KB) streams from the 192 MB L2, shared by all 9216 batches.
// 1 WG = 21 waves (one 16-row tile of N=325 each) x NB=2 batches; one
// workgroup barrier per Chebyshev order.

typedef __attribute__((ext_vector_type(2))) float v2f;
typedef __attribute__((ext_vector_type(8))) float v8f;

#define NROW 325
#define CIN  16
#define NCO  8
#define KORD 5
#define NB   2
#define NWAVE 21
#define NTHR (NWAVE * 32)

static __device__ __forceinline__ v8f wmma4(v2f a, v2f b, v8f c) {
  // D = A(16x4,f32) x B(4x16,f32) + C(16x16,f32)
  return __builtin_amdgcn_wmma_f32_16x16x4_f32(false, a, false, b, (short)0, c,
                                               false, false);
}

__global__ __launch_bounds__(NTHR) void cheb_fused(
    const float* __restrict__ x,    // [BT, 325, 16]
    const float* __restrict__ L,    // [325, 325] symmetric
    const float* __restrict__ W,    // [5, 16, 8]
    const float* __restrict__ bias, // [8]
    float* __restrict__ out,        // [BT, 325, 8]
    int BT)
{
  __shared__ __align__(16) float zbuf[2][NB][NROW * CIN];

  const int tid  = threadIdx.x;
  const int wave = tid >> 5;        // 0..20 : row tile of N
  const int lane = tid & 31;
  const int lh   = lane >> 4;       // lane half (K offset 0 or 2)
  const int l16  = lane & 15;       // column / M index within fragment
  const int batch0 = blockIdx.x * NB;

  // ---- W_k as WMMA B-fragments: K = c (16), N = co padded 8->16 w/ zeros ----
  v2f wb[KORD][4];
#pragma unroll
  for (int k = 0; k < KORD; ++k)
#pragma unroll
    for (int s = 0; s < 4; ++s) {
      int c0 = s * 4 + lh * 2;
      float w0 = 0.f, w1 = 0.f;
      if (l16 < NCO) {
        w0 = W[(k * CIN + c0    ) * NCO + l16];
        w1 = W[(k * CIN + c0 + 1) * NCO + l16];
      }
      wb[k][s].x = w0; wb[k][s].y = w1;
    }
  const float bs = (l16 < NCO) ? bias[l16] : 0.f;

  // ---- cooperative load: Z0 = x[batch] -> zbuf[0][nb] (float4 coalesced) ----
#pragma unroll
  for (int nb = 0; nb < NB; ++nb) {
    int b = batch0 + nb; if (b > BT - 1) b = BT - 1;
    const float4* src = (const float4*)(x + (size_t)b * (NROW * CIN));
    float4* dst = (float4*)&zbuf[0][nb][0];
    for (int i = tid; i < (NROW * CIN) / 4; i += NTHR) dst[i] = src[i];
  }
  __syncthreads();

  // ---- lane-fixed indices (rows clamped for the partial tile, wave 20) ----
  int rA = wave * 16 + l16; if (rA > NROW - 1) rA = NROW - 1; // A-matrix row
  const float* Lp    = L + (size_t)rA * NROW + lh * 2;        // main-loop ptr
  const float  Ltail = L[(size_t)rA * NROW + (NROW - 1)];     // K = 324
  const int    mA    = rA;                                    // proj A rows
  const int    rbase = wave * 16 + lh * 8;                    // C/D row base

  v8f oacc[NB] = {{0.f,0.f,0.f,0.f,0.f,0.f,0.f,0.f},
                  {0.f,0.f,0.f,0.f,0.f,0.f,0.f,0.f}};

  // ---- out += Z0 @ W0 ----
#pragma unroll
  for (int nb = 0; nb < NB; ++nb) {
    const float* zk = &zbuf[0][nb][0];
#pragma unroll
    for (int s = 0; s < 4; ++s) {
      int ai = mA * CIN + s * 4 + lh * 2;
      v2f a; a.x = zk[ai]; a.y = zk[ai + 1];
      oacc[nb] = wmma4(a, wb[0][s], oacc[nb]);
    }
  }

  // ---- Chebyshev recurrence, ping-pong buffers ----
  int cur = 0;
  for (int k = 1; k < KORD; ++k) {
    if (k >= 2) __syncthreads();    // all waves finished writing zbuf[cur]

    v8f y[NB] = {{0.f,0.f,0.f,0.f,0.f,0.f,0.f,0.f},
                 {0.f,0.f,0.f,0.f,0.f,0.f,0.f,0.f}};
    const float* z0 = &zbuf[cur][0][0];
    const float* z1 = &zbuf[cur][1][0];

    // Y = L x Zcur : 81 full K-steps (K = 0..323)
    for (int k0 = 0; k0 < NROW - 1; k0 += 4) {
      v2f a; a.x = Lp[k0]; a.y = Lp[k0 + 1];
      int zi = (k0 + lh * 2) * CIN + l16;
      v2f b0; b0.x = z0[zi]; b0.y = z0[zi + CIN];
      y[0] = wmma4(a, b0, y[0]);
      v2f b1; b1.x = z1[zi]; b1.y = z1[zi + CIN];
      y[1] = wmma4(a, b1, y[1]);
    }
    { // exact tail, K = 324 only (upper fragment halves zeroed)
      v2f a; a.x = lh ? 0.f : Ltail; a.y = 0.f;
      int zi = (NROW - 1) * CIN + l16;
      v2f b0; b0.x = lh ? 0.f : z0[zi]; b0.y = 0.f;
      y[0] = wmma4(a, b0, y[0]);
      v2f b1; b1.x = lh ? 0.f : z1[zi]; b1.y = 0.f;
      y[1] = wmma4(a, b1, y[1]);
    }

    // Z_k = (k==1) ? Y : 2Y - Z_{k-2}; overwrite the other buffer (own rows
    // only -> no cross-wave hazard), then out += Z_k @ W_k from LDS.
#pragma unroll
    for (int nb = 0; nb < NB; ++nb) {
      float* zn = &zbuf[cur ^ 1][nb][0];
#pragma unroll
      for (int v = 0; v < 8; ++v) {
        int r = rbase + v;
        if (r < NROW) {
          int idx = r * CIN + l16;
          float val = (k == 1) ? y[nb][v] : (2.f * y[nb][v] - zn[idx]);
          zn[idx] = val;
        }
      }
#pragma unroll
      for (int s = 0; s < 4; ++s) {
        int ai = mA * CIN + s * 4 + lh * 2;
        v2f a; a.x = zn[ai]; a.y = zn[ai + 1];
        oacc[nb] = wmma4(a, wb[k][s], oacc[nb]);
      }
    }
    cur ^= 1;
  }

  // ---- store out (+bias); C/D layout: row = rbase+v, col = l16 ----
#pragma unroll
  for (int nb = 0; nb < NB; ++nb) {
    int b = batch0 + nb;
    if (b < BT) {
      float* op = out + (size_t)b * NROW * NCO;
#pragma unroll
      for (int v = 0; v < 8; ++v) {
        int r = rbase + v;
        if (r < NROW && l16 < NCO)
          op[r * NCO + l16] = oacc[nb][v] + bs;
      }
    }
  }
}

extern "C" void kernel_launch(void* const* d_in, const int* in_sizes, int n_in,
                              void* d_out, int out_size, void* d_ws, size_t ws_size,
                              hipStream_t stream) {
  const float* x    = (const float*)d_in[0];
  const float* L    = (const float*)d_in[1];
  const float* W    = (const float*)d_in[2];
  const float* bias = (const float*)d_in[3];
  float* out = (float*)d_out;

  const int BT   = in_sizes[0] / (NROW * CIN);   // 9216
  const int nblk = (BT + NB - 1) / NB;           // 4608 workgroups

  cheb_fused<<<dim3(nblk), dim3(NTHR), 0, stream>>>(x, L, W, bias, out, BT);

  (void)n_in; (void)out_size; (void)d_ws; (void)ws_size;
}